// SobelKernel_51651276702241
// MI455X (gfx1250) — compile-verified
//
#include <hip/hip_runtime.h>
#include <math.h>

// MI455X fused mean -> 17-tap separable gaussian (reflect) -> sobel (zero pad) -> hypot.
// Gaussian convolutions run on the matrix pipe as banded-matrix WMMAs
// (V_WMMA_F32_16X16X4_F32, K accumulated in 8 chunks of 4), full fp32.

typedef float v2f __attribute__((ext_vector_type(2)));
typedef float v8f __attribute__((ext_vector_type(8)));

#define IMG 512
#define TILE 64
#define GS_R 82          // 64 + 2*(8+1) halo rows of channel-mean
#define GS_C 82
#define GS_STRIDE 84
#define HB_R 66          // 64 + 2*1 (sobel halo) rows of H-blurred data
#define HB_STRIDE 84
#define BL_STRIDE 68

__device__ __forceinline__ int reflect512(int i) {
    if (i < 0) return -i - 1;          // jnp 'symmetric' (edge-including reflect)
    if (i >= IMG) return 2 * IMG - 1 - i;
    return i;
}

__global__ __launch_bounds__(256) void sobel_fused(const float* __restrict__ x,
                                                   float* __restrict__ out) {
    // gs: 82x84 = 6888 f  | hb: 66x84 = 5544 f | bl (aliases gs): 66x68 = 4488 f | lgw: 17 f
    __shared__ float smem[6888 + 5544 + 17];
    float* gs  = smem;
    float* hb  = smem + 6888;
    float* bl  = smem;                 // reuse gs storage after H-blur
    float* lgw = smem + 6888 + 5544;

    const int tid  = threadIdx.x;
    const int lane = tid & 31;
    const int wave = tid >> 5;
    const int h0 = blockIdx.y * TILE;
    const int w0 = blockIdx.x * TILE;
    const int b  = blockIdx.z;

    // --- gaussian weights (sigma=2, radius=8), normalized, in LDS ---
    if (tid < 17) {
        float d = (float)(tid - 8);
        lgw[tid] = expf(-0.125f * d * d);
    }
    __syncthreads();
    float wsum = 0.f;
#pragma unroll
    for (int i = 0; i < 17; ++i) wsum += lgw[i];
    const float winv = 1.0f / wsum;

    // --- per-lane banded weight fragment (A layout for H pass, B layout for W pass;
    //     identical values since both index wgt[k - (lane&15)] with the same K split) ---
    const int mn    = lane & 15;
    const int khalf = (lane >> 4) * 2;
    v2f wf[8];
#pragma unroll
    for (int kc = 0; kc < 8; ++kc) {
        int j0 = kc * 4 + khalf - mn;
        int j1 = j0 + 1;
        int j0c = min(max(j0, 0), 16);
        int j1c = min(max(j1, 0), 16);
        wf[kc].x = (j0 >= 0 && j0 <= 16) ? lgw[j0c] * winv : 0.f;
        wf[kc].y = (j1 >= 0 && j1 <= 16) ? lgw[j1c] * winv : 0.f;
    }

    // --- phase 1: channel mean with reflect halo -> gs[82][82] ---
    const float* xb = x + (size_t)b * 3 * IMG * IMG;
    for (int i = tid; i < GS_R * GS_C; i += 256) {
        int r = i / GS_C;
        int c = i - r * GS_C;
        int gh = reflect512(h0 - 9 + r);
        int gw = reflect512(w0 - 9 + c);
        size_t base = (size_t)gh * IMG + gw;
        float v0 = xb[base];
        float v1 = xb[base + IMG * IMG];
        float v2 = xb[base + 2 * IMG * IMG];
        gs[r * GS_STRIDE + c] = (v0 + v1 + v2) * (1.0f / 3.0f);
    }
    __syncthreads();

    // --- phase 2: H-blur via WMMA: hb[r][c] = sum_k wgt[k-m] * gs[R+k][c] ---
    // row tiles {0,16,32,48,50} cover 0..65; col tiles {0,16,32,48,64,66} cover 0..81
    for (int t = wave; t < 30; t += 8) {
        int R = min((t / 6) * 16, 50);
        int C = (t % 6) < 5 ? min((t % 6) * 16, 64) : 66;
        v8f acc = {};
#pragma unroll
        for (int kc = 0; kc < 8; ++kc) {
            int row0 = R + kc * 4 + khalf;
            v2f bfrag;
            bfrag.x = gs[row0 * GS_STRIDE + C + mn];
            bfrag.y = gs[(row0 + 1) * GS_STRIDE + C + mn];
            acc = __builtin_amdgcn_wmma_f32_16x16x4_f32(
                false, wf[kc], false, bfrag, (short)0, acc, false, false);
        }
        int mrow = R + (lane >> 4) * 8;
        int ncol = C + mn;
#pragma unroll
        for (int v = 0; v < 8; ++v)
            hb[(mrow + v) * HB_STRIDE + ncol] = acc[v];
    }
    __syncthreads();

    // --- phase 3: W-blur via WMMA: bl[r][c] = sum_k hb[r][Cb+k] * wgt[k-n];
    //     zero-masked outside the image (sobel zero padding) ---
    for (int t = wave; t < 25; t += 8) {
        int R  = min((t / 5) * 16, 50);
        int Cb = min((t % 5) * 16, 50);
        v8f acc = {};
#pragma unroll
        for (int kc = 0; kc < 8; ++kc) {
            int col0 = Cb + kc * 4 + khalf;
            v2f afrag;
            afrag.x = hb[(R + mn) * HB_STRIDE + col0];
            afrag.y = hb[(R + mn) * HB_STRIDE + col0 + 1];
            acc = __builtin_amdgcn_wmma_f32_16x16x4_f32(
                false, afrag, false, wf[kc], (short)0, acc, false, false);
        }
        int mrow = R + (lane >> 4) * 8;
        int ncol = Cb + mn;
        int iw = w0 - 1 + ncol;
        bool cok = (iw >= 0) && (iw < IMG);
#pragma unroll
        for (int v = 0; v < 8; ++v) {
            int r = mrow + v;
            int ih = h0 - 1 + r;
            float val = (cok && ih >= 0 && ih < IMG) ? acc[v] : 0.f;
            bl[r * BL_STRIDE + ncol] = val;
        }
    }
    __syncthreads();

    // --- phase 4: sobel + hypot ---
    float* ob = out + ((size_t)b * IMG + h0) * IMG + w0;
    for (int i = tid; i < TILE * TILE; i += 256) {
        int r = i >> 6;
        int c = i & 63;
        int br = r + 1, bc = c + 1;
        float tl = bl[(br - 1) * BL_STRIDE + (bc - 1)];
        float tc = bl[(br - 1) * BL_STRIDE + bc];
        float tr = bl[(br - 1) * BL_STRIDE + (bc + 1)];
        float ml = bl[br * BL_STRIDE + (bc - 1)];
        float mr = bl[br * BL_STRIDE + (bc + 1)];
        float dl = bl[(br + 1) * BL_STRIDE + (bc - 1)];
        float dc = bl[(br + 1) * BL_STRIDE + bc];
        float dr = bl[(br + 1) * BL_STRIDE + (bc + 1)];
        // sx: d/dH ([-1,0,1]) then [1,2,1] along W ; sy: d/dW then [1,2,1] along H
        float sx = (dl + 2.f * dc + dr) - (tl + 2.f * tc + tr);
        float sy = (tr + 2.f * mr + dr) - (tl + 2.f * ml + dl);
        ob[(size_t)r * IMG + c] = sqrtf(sx * sx + sy * sy);
    }
}

extern "C" void kernel_launch(void* const* d_in, const int* in_sizes, int n_in,
                              void* d_out, int out_size, void* d_ws, size_t ws_size,
                              hipStream_t stream) {
    (void)in_sizes; (void)n_in; (void)out_size; (void)d_ws; (void)ws_size;
    const float* x = (const float*)d_in[0];
    float* out = (float*)d_out;
    dim3 grid(IMG / TILE, IMG / TILE, 32);   // (8, 8, 32)
    sobel_fused<<<grid, 256, 0, stream>>>(x, out);
}